// GP_REIM_8881992368205
// MI455X (gfx1250) — compile-verified
//
#include <hip/hip_runtime.h>

typedef float v2f __attribute__((ext_vector_type(2)));
typedef float v8f __attribute__((ext_vector_type(8)));

#define COLSPLIT 8
#define LOG2E 1.44269504088896340736f

// ds_swizzle group-of-32: offset = xor_mask<<10 | or_mask<<5 | and_mask
#define SWZ_ADD(v, pat) \
    ((v) + __int_as_float(__builtin_amdgcn_ds_swizzle(__float_as_int(v), (pat))))

__global__ void GP_zero_kernel(float* out, int n) {
    int i = blockIdx.x * blockDim.x + threadIdx.x;
    if (i < n) out[i] = 0.0f;
}

// One wave handles a 16-row test tile over N/COLSPLIT train points.
// t[i][j] = cexp * ( -2*x_i.x_j + |x_i|^2 + |x_j|^2 ) computed as a single
// rank-4 product via V_WMMA_F32_16X16X4_F32, with cexp pre-folded into A
// (K=2 coords + 2 norm columns). Then K = clamp01(exp2(t)) — the clamp is
// equivalent to the reference's max(sq,0) (exp2 is monotone, cexp<0) and
// folds into the v_exp_f32 destination clamp modifier.
__global__ __launch_bounds__(256) void GP_mean_kernel(
    const float2* __restrict__ Xte,   // (M,2)
    const float2* __restrict__ Xtr,   // (N,2)
    const float2* __restrict__ Alp,   // (N,2) [Re, Im]
    const float* __restrict__ p_ll,   // log_lengthscale (scalar)
    const float* __restrict__ p_lo,   // log_outputscale (scalar)
    float* __restrict__ out,          // (M,2)
    int N)
{
    const int lane  = threadIdx.x & 31;
    const int r16   = lane & 15;
    const bool lo16 = (lane < 16);

    // wave-uniform values -> force into SGPRs so the train loop is scalar
    const int gwave = __builtin_amdgcn_readfirstlane(
        blockIdx.x * (blockDim.x >> 5) + (threadIdx.x >> 5));
    const int rowtile = gwave / COLSPLIT;
    const int colseg  = gwave % COLSPLIT;
    const int nseg    = __builtin_amdgcn_readfirstlane(N / COLSPLIT);
    const int nbase0  = colseg * nseg;

    // hyper-params: ell2 = exp(2*ll), sf2 = exp(2*lo)
    const float ll  = p_ll[0];
    const float lo  = p_lo[0];
    const float inv_ell2 = __builtin_amdgcn_exp2f(-2.0f * ll * LOG2E); // exp(-2*ll)
    const float cexp = -0.5f * inv_ell2 * LOG2E;   // K = exp2(cexp * sq), cexp < 0
    const float sf2  = __builtin_amdgcn_exp2f(2.0f * lo * LOG2E);

    // A matrix (16x4 f32, 2 VGPRs): lanes 0-15 = K0,K1 ; lanes 16-31 = K2,K3
    //   row i: cexp * [ -2*x, -2*y, |x|^2, 1 ]   (cexp folded in once)
    const float2 xt = Xte[rowtile * 16 + r16];
    const float xtsq = xt.x * xt.x + xt.y * xt.y;
    v2f A;
    A.x = cexp * (lo16 ? (-2.0f * xt.x) : xtsq);   // K=0 | K=2
    A.y = cexp * (lo16 ? (-2.0f * xt.y) : 1.0f);   // K=1 | K=3

    float accre[8], accim[8];
#pragma unroll
    for (int r = 0; r < 8; ++r) { accre[r] = 0.0f; accim[r] = 0.0f; }

#pragma unroll 2
    for (int nb = 0; nb < nseg; nb += 16) {
        const int col = nbase0 + nb + r16;
        const float2 tr = Xtr[col];
        const float2 al = Alp[col];

        // B matrix (4x16 f32, 2 VGPRs): lanes 0-15 = K0,K1 rows; 16-31 = K2,K3
        //   col j: [ x', y', 1, |x'|^2 ]^T
        const float trsq = tr.x * tr.x + tr.y * tr.y;
        v2f B;
        B.x = lo16 ? tr.x : 1.0f;   // K=0 row | K=2 row (ones)
        B.y = lo16 ? tr.y : trsq;   // K=1 row | K=3 row (|x'|^2)

        v8f C = {0.f, 0.f, 0.f, 0.f, 0.f, 0.f, 0.f, 0.f};
        // t tile = cexp*sq: element (M = r + 8*(lane>=16), N = lane&15)
        v8f S = __builtin_amdgcn_wmma_f32_16x16x4_f32(
            false, A, false, B, (short)0, C, false, false);

#pragma unroll
        for (int r = 0; r < 8; ++r) {
            // K = min(exp2(t), 1) == exp2(min(t,0)); med3(k,0,1) folds into
            // the clamp destination modifier of v_exp_f32 (k is canonical,
            // k >= 0), so the reference's max(sq,0) costs zero instructions.
            float k = __builtin_amdgcn_exp2f(S[r]);
            k = __builtin_amdgcn_fmed3f(k, 0.0f, 1.0f);
            accre[r] = fmaf(k, al.x, accre[r]);
            accim[r] = fmaf(k, al.y, accim[r]);
        }
    }

    // Reduce over the 16 train columns held across each 16-lane half
    // (ds_swizzle SWAPX1/2/4/8 stays within each half).
#pragma unroll
    for (int r = 0; r < 8; ++r) {
        float re = accre[r], im = accim[r];
        re = SWZ_ADD(re, 0x041f);  im = SWZ_ADD(im, 0x041f);   // xor 1
        re = SWZ_ADD(re, 0x081f);  im = SWZ_ADD(im, 0x081f);   // xor 2
        re = SWZ_ADD(re, 0x101f);  im = SWZ_ADD(im, 0x101f);   // xor 4
        re = SWZ_ADD(re, 0x201f);  im = SWZ_ADD(im, 0x201f);   // xor 8
        accre[r] = re; accim[r] = im;
    }

    // lane 0 owns rows 0-7 of the tile, lane 16 owns rows 8-15
    if (r16 == 0) {
        const int baserow = rowtile * 16 + (lane >> 4) * 8;
#pragma unroll
        for (int r = 0; r < 8; ++r) {
            atomicAdd(&out[(baserow + r) * 2 + 0], sf2 * accre[r]);
            atomicAdd(&out[(baserow + r) * 2 + 1], sf2 * accim[r]);
        }
    }
}

extern "C" void kernel_launch(void* const* d_in, const int* in_sizes, int n_in,
                              void* d_out, int out_size, void* d_ws, size_t ws_size,
                              hipStream_t stream) {
    const float2* Xte = (const float2*)d_in[0];
    const float2* Xtr = (const float2*)d_in[1];
    const float2* Alp = (const float2*)d_in[2];
    const float*  ll  = (const float*)d_in[3];
    const float*  lo  = (const float*)d_in[4];
    float* out = (float*)d_out;

    const int M = in_sizes[0] / 2;
    const int N = in_sizes[1] / 2;

    GP_zero_kernel<<<(out_size + 255) / 256, 256, 0, stream>>>(out, out_size);

    const int waves   = (M / 16) * COLSPLIT;   // 8192 waves
    const int threads = waves * 32;
    GP_mean_kernel<<<threads / 256, 256, 0, stream>>>(Xte, Xtr, Alp, ll, lo, out, N);
}